// SpectralBucketLM_56882546868555
// MI455X (gfx1250) — compile-verified
//
#include <hip/hip_runtime.h>

// Problem constants (from reference setup_inputs)
#define SPAN 8
#define B_ 16
#define T_ 8192
#define D_ 192
#define K_ 512
#define S_ (T_ / SPAN)   // 1024 segments per batch row
#define M_ (B_ * S_)     // 16384 total segments
#define NT_ (K_ / 16)    // 32 code tiles

// Output layout (floats, concatenated in reference return order)
#define Q_OFF   0
#define LOSS_OFF ((long)B_ * T_ * D_)          // 25,165,824
#define IDX_OFF  (LOSS_OFF + 1)
#define BND_OFF  (IDX_OFF + (long)B_ * T_)

// LDS tile: 16 codebook rows, padded to 196 floats (784 B, 16B aligned) so
// B-fragment ds_load_b64 addresses hit bank (4*ln + c) % 64 -> conflict-free.
#define ROWPAD 196

typedef __attribute__((ext_vector_type(2))) float v2f;
typedef __attribute__((ext_vector_type(8))) float v8f;

// CDNA5 async global->LDS copy (ASYNCcnt-tracked, no VGPR staging).
__device__ __forceinline__ void async_ld16(unsigned lds_off, unsigned long long gaddr) {
    asm volatile("global_load_async_to_lds_b128 %0, %1, off"
                 :: "v"(lds_off), "v"(gaddr) : "memory");
}

// ---------------------------------------------------------------------------
// Kernel 1: per-code 0.5*||c||^2 + zero loss accumulators (stream-ordered init)
// ---------------------------------------------------------------------------
__global__ void k_prep(const float* __restrict__ codebook,
                       float* __restrict__ cnorm,
                       float* __restrict__ accum) {
    int k = blockIdx.x * blockDim.x + threadIdx.x;
    if (k == 0) { accum[0] = 0.f; accum[1] = 0.f; }
    if (k < K_) {
        float s = 0.f;
#pragma unroll 4
        for (int d = 0; d < D_; ++d) {
            float c = codebook[(long)k * D_ + d];
            s += c * c;
        }
        cnorm[k] = 0.5f * s;
    }
}

// ---------------------------------------------------------------------------
// Kernel 2: segment pooling + boundary predictor.
// One wave32 per segment (8 tokens x 192 dims); lane owns 6 strided dims.
// ---------------------------------------------------------------------------
__global__ void k_pool_boundary(const float* __restrict__ x,
                                const float* __restrict__ bw,
                                const float* __restrict__ bbias,
                                float* __restrict__ pooled,
                                float* __restrict__ bnd_out,
                                float* __restrict__ accum) {
    const int lane = threadIdx.x & 31;
    const int wid  = threadIdx.x >> 5;
    const int m = blockIdx.x * 8 + wid;     // segment id in [0, M_)
    const int b = m / S_;
    const int s = m - b * S_;
    const long xbase = ((long)b * T_ + (long)s * SPAN) * D_;

    float wv[6], ps[6];
#pragma unroll
    for (int q = 0; q < 6; ++q) { wv[q] = bw[lane + 32 * q]; ps[q] = 0.f; }
    const float bias = bbias[0];
    float cnt = 0.f;

    for (int jt = 0; jt < SPAN; ++jt) {
        float part = 0.f;
#pragma unroll
        for (int q = 0; q < 6; ++q) {
            float v = x[xbase + (long)jt * D_ + lane + 32 * q];
            ps[q] += v;
            part  += v * wv[q];
        }
#pragma unroll
        for (int off = 16; off >= 1; off >>= 1)
            part += __shfl_xor(part, off, 32);
        // sigmoid(logit) > 0.5  <=>  logit > 0
        float bval = (part + bias) > 0.f ? 1.f : 0.f;
        if (lane == 0) {
            bnd_out[(long)b * T_ + (long)s * SPAN + jt] = bval;
            cnt += bval;
        }
    }
#pragma unroll
    for (int q = 0; q < 6; ++q)
        pooled[(long)m * D_ + lane + 32 * q] = ps[q] * (1.f / SPAN);
    if (lane == 0) atomicAdd(&accum[1], cnt);
}

// ---------------------------------------------------------------------------
// Kernel 3: VQ nearest-code search via f32 WMMA.
// score_k = p.c_k - 0.5||c_k||^2 ; argmin d2 == argmax score.
// Codebook tiles double-buffered in LDS via async global->LDS copies;
// one wave per 16-row tile; 48 x v_wmma_f32_16x16x4_f32 per 16x16 score tile.
// ---------------------------------------------------------------------------
__global__ void k_vq_wmma(const float* __restrict__ pooled,
                          const float* __restrict__ codebook,
                          const float* __restrict__ cnorm,
                          int* __restrict__ segidx) {
    __shared__ __align__(16) float smem[2][16 * ROWPAD];

    const unsigned tid = threadIdx.x;
    const int lane = tid & 31;
    const int wid  = tid >> 5;
    const int half = lane >> 4;     // 0: K={0,1} ; 1: K={2,3}
    const int ln   = lane & 15;
    const int row0 = (blockIdx.x * 8 + wid) * 16;

    // Preload all A fragments for this 16-row tile (reused across 32 n-tiles).
    const float* pA = pooled + (long)(row0 + ln) * D_ + 2 * half;
    v2f areg[48];
#pragma unroll
    for (int kk = 0; kk < 48; ++kk) areg[kk] = *(const v2f*)(pA + 4 * kk);

    const unsigned lds0 = (unsigned)(size_t)(void*)&smem[0][0];
    const unsigned lds1 = (unsigned)(size_t)(void*)&smem[1][0];

    // Stage one 16x192 codebook tile into LDS buffer `bsel`.
    // 768 16B-chunks / 256 threads = 3 async b128 ops per wave.
    auto stage = [&](int n0, int bsel) {
        unsigned base = bsel ? lds1 : lds0;
#pragma unroll
        for (int r = 0; r < 3; ++r) {
            int idx    = (int)tid + r * 256;        // chunk 0..767
            int row    = idx / 48;                  // 48 chunks per row
            int within = (idx - row * 48) * 16;     // bytes within row
            unsigned long long g = (unsigned long long)(size_t)codebook
                                 + (unsigned long long)(n0 + row) * (D_ * 4) + within;
            async_ld16(base + (unsigned)(row * (ROWPAD * 4) + within), g);
        }
    };

    float best[8];
    int   bidx[8];
#pragma unroll
    for (int r = 0; r < 8; ++r) { best[r] = -3.0e38f; bidx[r] = 0; }

    stage(0, 0);
#pragma unroll 1
    for (int t = 0; t < NT_; ++t) {
        if (t + 1 < NT_) {
            stage((t + 1) * 16, (t + 1) & 1);
            // 3 newer async ops may remain in flight; older tile is complete.
            asm volatile("s_wait_asynccnt 0x3" ::: "memory");
        } else {
            asm volatile("s_wait_asynccnt 0x0" ::: "memory");
        }
        __syncthreads();            // tile t visible to all waves

        const float* tile = &smem[t & 1][0];
        const int lrow = ln * ROWPAD + 2 * half;

        v8f acc = {};
#pragma unroll
        for (int g = 0; g < 4; ++g) {
            v2f bf[12];
#pragma unroll
            for (int i = 0; i < 12; ++i)
                bf[i] = *(const v2f*)(tile + lrow + 4 * (g * 12 + i));
#pragma unroll
            for (int i = 0; i < 12; ++i)
                // (neg_a, A, neg_b, B, c_mod, C, reuse_a, reuse_b)
                acc = __builtin_amdgcn_wmma_f32_16x16x4_f32(
                    false, areg[g * 12 + i], false, bf[i], (short)0, acc,
                    false, false);
        }

        const float cn  = cnorm[t * 16 + ln];
        const int  code = t * 16 + ln;
#pragma unroll
        for (int r = 0; r < 8; ++r) {
            float sc = acc[r] - cn;   // lanes0-15: row r ; lanes16-31: row r+8
            if (sc > best[r] || (sc == best[r] && code < bidx[r])) {
                best[r] = sc; bidx[r] = code;
            }
        }
        __syncthreads();            // all waves done reading buf[t&1] before
                                    // it is restaged at iteration t+1
    }

    // Reduce across the 16 lanes of each half (xor 8..1 stays within a half)
#pragma unroll
    for (int r = 0; r < 8; ++r) {
        float bv = best[r];
        int   bi = bidx[r];
#pragma unroll
        for (int off = 8; off >= 1; off >>= 1) {
            float ov = __shfl_xor(bv, off, 32);
            int   oi = __shfl_xor(bi, off, 32);
            if (ov > bv || (ov == bv && oi < bi)) { bv = ov; bi = oi; }
        }
        if (ln == 0) segidx[row0 + r + 8 * half] = bi;
    }
}

// ---------------------------------------------------------------------------
// Kernel 4: expand quantized segments to token resolution + commitment loss.
// One block (192 threads) per segment; writes 8 tokens of q_out.
// ---------------------------------------------------------------------------
__global__ void k_expand(const float* __restrict__ codebook,
                         const float* __restrict__ pooled,
                         const int* __restrict__ segidx,
                         float* __restrict__ out,
                         float* __restrict__ accum) {
    const int m = blockIdx.x;
    const int d = threadIdx.x;          // 0..191
    const int idx = segidx[m];
    const float c = codebook[(long)idx * D_ + d];
    const float p = pooled[(long)m * D_ + d];
    float diff = c - p;
    float sq = diff * diff;
#pragma unroll
    for (int off = 16; off >= 1; off >>= 1)
        sq += __shfl_xor(sq, off, 32);
    if ((d & 31) == 0) atomicAdd(&accum[0], sq);

    const int b = m / S_;
    const int s = m - b * S_;
    const long tok0 = (long)b * T_ + (long)s * SPAN;
    float* q = out + Q_OFF + tok0 * D_ + d;
#pragma unroll
    for (int j = 0; j < SPAN; ++j)
        q[(long)j * D_] = c;            // STE output == quantized numerically
    if (d < SPAN)
        out[IDX_OFF + tok0 + d] = (float)idx;
}

// ---------------------------------------------------------------------------
// Kernel 5: scalar total loss
// ---------------------------------------------------------------------------
__global__ void k_loss(const float* __restrict__ accum, float* __restrict__ out) {
    float sq_mean   = accum[0] / (float)((long)M_ * D_);
    float ones_mean = accum[1] / (float)((long)B_ * T_);
    float br = ones_mean - (1.f / SPAN);
    out[LOSS_OFF] = 0.25f * sq_mean + 0.01f * br * br;
}

// ---------------------------------------------------------------------------
extern "C" void kernel_launch(void* const* d_in, const int* in_sizes, int n_in,
                              void* d_out, int out_size, void* d_ws, size_t ws_size,
                              hipStream_t stream) {
    const float* x        = (const float*)d_in[0];
    const float* codebook = (const float*)d_in[1];
    const float* bw       = (const float*)d_in[2];
    const float* bb       = (const float*)d_in[3];
    float* out = (float*)d_out;

    // Workspace layout
    float* pooled = (float*)d_ws;                    // M_*D_ floats (12.6 MB)
    float* cnorm  = pooled + (size_t)M_ * D_;        // K_ floats
    int*   segidx = (int*)(cnorm + K_);              // M_ ints
    float* accum  = (float*)(segidx + M_);           // 2 floats

    k_prep<<<2, 256, 0, stream>>>(codebook, cnorm, accum);
    k_pool_boundary<<<M_ / 8, 256, 0, stream>>>(x, bw, bb, pooled,
                                                out + BND_OFF, accum);
    k_vq_wmma<<<M_ / 16 / 8, 256, 0, stream>>>(pooled, codebook, cnorm, segidx);
    k_expand<<<M_, D_, 0, stream>>>(codebook, pooled, segidx, out, accum);
    k_loss<<<1, 1, 0, stream>>>(accum, out);
}